// MetropolisHastings_15058155340471
// MI455X (gfx1250) — compile-verified
//
#include <hip/hip_runtime.h>
#include <stdint.h>

// ---- problem constants (match reference) ----
#define T_TOTAL     1100000
#define NSAMP       1000000
#define BURNIN_N    100000
#define NOBS        256
#define INV_VAR     4.0f      // 1/(0.5*0.5)
#define IDEAL_RATE  0.234f
#define STEP0       0.1f
#define ULIMF       1.0f

// ---- chunking ----
#define CHUNK       1000                  // 1100 chunks; burn-in = exactly 100 chunks
#define NCHUNKS     (T_TOTAL / CHUNK)     // 1100
#define BURN_CHUNKS (BURNIN_N / CHUNK)    // 100

// ---- LDS layout (float indices; all 16B-aligned byte offsets) ----
#define EPS_A   0                 // CHUNK*2 floats (8000 B)
#define EPS_B   (CHUNK * 2)       // 2000
#define U_A     (CHUNK * 4)       // 4000
#define U_B     (CHUNK * 5)       // 5000
#define SAMP_A  (CHUNK * 6)       // 6000
#define SAMP_B  (CHUNK * 8)       // 8000
#define SMEM_FLOATS (CHUNK * 10)  // 10000 floats = 40 KB

typedef float v2f __attribute__((ext_vector_type(2)));
typedef float v8f __attribute__((ext_vector_type(8)));

// Per-lane 16B async copy global -> LDS (ASYNCcnt-tracked). One wave
// instruction moves 32 lanes x 16B = 512B; per-wave ASYNCcnt stays tiny.
__device__ __forceinline__ void async_copy_to_lds(const char* gsrc, unsigned ldsDst,
                                                  int nbytes, int tid) {
  for (int off = tid * 16; off < nbytes; off += 256 * 16) {
    unsigned l = ldsDst + (unsigned)off;
    const char* g = gsrc + off;
    asm volatile("global_load_async_to_lds_b128 %0, %1, off"
                 :: "v"(l), "v"(g)
                 : "memory");
  }
}

// Per-lane 16B async store LDS -> global (ASYNCcnt-tracked, fire-and-forget).
__device__ __forceinline__ void async_store_from_lds(char* gdst, unsigned ldsSrc,
                                                     int nbytes, int tid) {
  for (int off = tid * 16; off < nbytes; off += 256 * 16) {
    unsigned l = ldsSrc + (unsigned)off;
    char* g = gdst + off;
    asm volatile("global_store_async_from_lds_b128 %0, %1, off"
                 :: "v"(g), "v"(l)
                 : "memory");
  }
}

__device__ __forceinline__ void wait_async0() {
  asm volatile("s_wait_asynccnt 0" ::: "memory");
}

__global__ __launch_bounds__(256) void mh_chain_kernel(
    const float* __restrict__ xs,      // observation_locations [256]
    const float* __restrict__ ys,      // observations_values   [256]
    const float* __restrict__ theta0,  // [2]
    const float* __restrict__ eps,     // prop_noise [T,2]
    const float* __restrict__ ur,      // u_rand [T]
    float* __restrict__ out)           // samples [1e6,2] then acc_rate
{
  __shared__ __attribute__((aligned(16))) float smem[SMEM_FLOATS];
  const int tid = threadIdx.x;
  const unsigned ldsBase = (unsigned)(uintptr_t)&smem[0];  // low 32 bits = LDS offset

  // ---------------- Stage 1: sufficient statistics (one-time) ----------------
  // ll(t) = -0.5*inv_var * [ N t0^2 + Sxx t1^2 + 2 Sx t0 t1 - 2 Sy t0 - 2 Sxy t1 + Syy ]
  {
    float x = xs[tid];
    float y = ys[tid];
    smem[tid]         = x;      // stat row 0: Sx
    smem[256  + tid]  = y;      // stat row 1: Sy
    smem[512  + tid]  = x * x;  // stat row 2: Sxx
    smem[768  + tid]  = x * y;  // stat row 3: Sxy
    smem[1024 + tid]  = y * y;  // stat row 4: Syy
  }
  __syncthreads();
  // LDS tree down to 4 partials per statistic.
  for (int s = 128; s >= 4; s >>= 1) {
    if (tid < s) {
      smem[tid]        += smem[tid + s];
      smem[256  + tid] += smem[256  + tid + s];
      smem[512  + tid] += smem[512  + tid + s];
      smem[768  + tid] += smem[768  + tid + s];
      smem[1024 + tid] += smem[1024 + tid + s];
    }
    __syncthreads();
  }

  // Finish with one V_WMMA_F32_16X16X4_F32: D = A(16x4) * B(4x16), where
  // row m of A holds statistic m's 4 partials and B's column 0 is all-ones.
  // A layout (32-bit 16x4): lane m -> {K0,K1} in VGPR0/1; lane m+16 -> {K2,K3}.
  // D[m,0] lands in lane 0 (lanes 0-15 cover N=0..15), VGPR m, for m<8.
  {
    const unsigned lane = (unsigned)tid & 31u;
    const unsigned row  = lane & 15u;          // M for A, N for B
    const bool     hi   = lane >= 16u;         // K2/K3 half
    const unsigned rr   = (row < 5u) ? row : 0u;   // clamp: rows 5..15 zeroed
    float a0 = smem[256u * rr + (hi ? 2u : 0u)];
    float a1 = smem[256u * rr + (hi ? 3u : 1u)];
    a0 = (row < 5u) ? a0 : 0.0f;
    a1 = (row < 5u) ? a1 : 0.0f;
    const float bv = (row == 0u) ? 1.0f : 0.0f;    // column N=0 = ones (both K-halves)
    v2f Av = {a0, a1};
    v2f Bv = {bv, bv};
    v8f Cv = {};
    Cv = __builtin_amdgcn_wmma_f32_16x16x4_f32(
        /*neg_a=*/false, Av, /*neg_b=*/false, Bv,
        /*c_mod=*/(short)0, Cv, /*reuse_a=*/false, /*reuse_b=*/false);
    __syncthreads();                           // A reads done before smem[0..4] rewrite
    if (tid == 0) {                            // lane 0 of wave 0 holds D[m,0] in Cv[m]
      smem[0] = Cv[0];  // Sx
      smem[1] = Cv[1];  // Sy
      smem[2] = Cv[2];  // Sxx
      smem[3] = Cv[3];  // Sxy
      smem[4] = Cv[4];  // Syy
    }
  }
  __syncthreads();
  const float Sx  = smem[0];
  const float Sy  = smem[1];
  const float Sxx = smem[2];
  const float Sxy = smem[3];
  const float Syy = smem[4];
  __syncthreads();  // region reused as eps buffer below

  // lp(t0,t1) = A t0^2 + B t1^2 + C t0 t1 + D t0 + E t1 + F
  const float A = -0.5f * INV_VAR * (float)NOBS;
  const float B = -0.5f * INV_VAR * Sxx;
  const float C = -INV_VAR * Sx;
  const float D =  INV_VAR * Sy;
  const float E =  INV_VAR * Sxy;
  const float F = -0.5f * INV_VAR * Syy;

  float t0 = theta0[0];
  float t1 = theta0[1];
  float dt = STEP0;
  float fi = 0.0f;
  float lpc = fmaf(fmaf(A, t0, fmaf(C, t1, D)), t0, fmaf(fmaf(B, t1, E), t1, F));
  unsigned naccept = 0;

  // ---------------- Prologue: async-load chunk 0 into buffer A ----------------
  async_copy_to_lds((const char*)eps, ldsBase + 4u * EPS_A, CHUNK * 2 * 4, tid);
  async_copy_to_lds((const char*)ur,  ldsBase + 4u * U_A,   CHUNK * 4,     tid);
  wait_async0();
  __syncthreads();

  // ---------------- Main chunked sequential chain ----------------
  for (int c = 0; c < NCHUNKS; ++c) {
    const int cur = c & 1;
    const float2* __restrict__ e2 = (const float2*)&smem[cur ? EPS_B : EPS_A];
    const float*  __restrict__ ub = &smem[cur ? U_B : U_A];
    float2* __restrict__ sb = (float2*)&smem[cur ? SAMP_B : SAMP_A];

    // Async-prefetch next chunk's randoms into the other input buffers.
    if (c + 1 < NCHUNKS) {
      const int nxt = (c + 1) & 1;
      async_copy_to_lds((const char*)(eps + (size_t)(c + 1) * CHUNK * 2),
                        ldsBase + 4u * (nxt ? EPS_B : EPS_A), CHUNK * 2 * 4, tid);
      async_copy_to_lds((const char*)(ur + (size_t)(c + 1) * CHUNK),
                        ldsBase + 4u * (nxt ? U_B : U_A), CHUNK * 4, tid);
    }
    // Async-drain previous chunk's samples LDS -> HBM while we compute.
    if (c >= 1 && (c - 1) >= BURN_CHUNKS) {
      const int prv = (c - 1) & 1;
      async_store_from_lds((char*)(out + (size_t)(c - 1 - BURN_CHUNKS) * CHUNK * 2),
                           ldsBase + 4u * (prv ? SAMP_B : SAMP_A), CHUNK * 2 * 4, tid);
    }

    // Sequential chain over this chunk (all 256 threads redundantly; latency-bound,
    // so redundancy is free and keeps every wave in lockstep for the copies).
    #pragma unroll 4
    for (int j = 0; j < CHUNK; ++j) {
      const float2 e = e2[j];
      const float  u = ub[j];
      const float rcp_i = __builtin_amdgcn_rcpf(fi + 1.0f);  // chain-independent

      const float p0 = fmaf(dt, e.x, t0);
      const float p1 = fmaf(dt, e.y, t1);
      const float lpp = fmaf(fmaf(A, p0, fmaf(C, p1, D)), p0,
                             fmaf(fmaf(B, p1, E), p1, F));
      const bool  inb = fmaxf(fabsf(p0), fabsf(p1)) <= ULIMF;
      const float a   = inb ? fminf(__expf(lpp - lpc), 1.0f) : 0.0f;
      const bool  acc = u < a;

      t0  = acc ? p0  : t0;
      t1  = acc ? p1  : t1;
      lpc = acc ? lpp : lpc;
      naccept += acc ? 1u : 0u;
      dt = fmaf(dt * (a - IDEAL_RATE), rcp_i, dt);
      fi += 1.0f;

      // All lanes write the identical value to the same LDS slot: race-free,
      // avoids a per-iteration exec-mask branch.
      sb[j] = make_float2(t0, t1);
    }

    __syncthreads();   // sample buffer complete & visible
    wait_async0();     // next chunk resident; previous samples drained (buffer reusable)
    __syncthreads();
  }

  // Flush the final chunk's samples.
  {
    const int last = NCHUNKS - 1;
    const int prv = last & 1;
    async_store_from_lds((char*)(out + (size_t)(last - BURN_CHUNKS) * CHUNK * 2),
                         ldsBase + 4u * (prv ? SAMP_B : SAMP_A), CHUNK * 2 * 4, tid);
    wait_async0();
  }

  if (tid == 0) {
    out[(size_t)NSAMP * 2] = (float)naccept / (float)NSAMP;
  }
}

extern "C" void kernel_launch(void* const* d_in, const int* in_sizes, int n_in,
                              void* d_out, int out_size, void* d_ws, size_t ws_size,
                              hipStream_t stream) {
  const float* xs     = (const float*)d_in[0];
  const float* ys     = (const float*)d_in[1];
  const float* theta0 = (const float*)d_in[2];
  const float* eps    = (const float*)d_in[3];
  const float* ur     = (const float*)d_in[4];
  float* out          = (float*)d_out;

  hipLaunchKernelGGL(mh_chain_kernel, dim3(1), dim3(256), 0, stream,
                     xs, ys, theta0, eps, ur, out);
}